// Context_RGR_20718922235945
// MI455X (gfx1250) — compile-verified
//
#include <hip/hip_runtime.h>
#include <math.h>

// Problem constants (match reference)
#define B_ 128
#define N_ 65536
#define D_ 512
#define KTOP 5
#define M_KEEP 256          // D * RATIO
#define TILE_N 64
#define NUM_TILES (N_ / TILE_N)   // 1024

typedef __attribute__((ext_vector_type(2))) float v2f;
typedef __attribute__((ext_vector_type(8))) float v8f;

// ---------------------------------------------------------------------------
// 1) g_inv[n] = 1 / max(||gallery[n,:]||, 1e-12)   (one wave per row)
// ---------------------------------------------------------------------------
__global__ void k_ginv(const float* __restrict__ gal, float* __restrict__ g_inv) {
  const int wave = threadIdx.x >> 5;
  const int lane = threadIdx.x & 31;
  const int row  = blockIdx.x * 8 + wave;
  const float* p = gal + (size_t)row * D_;
  float ss = 0.f;
  #pragma unroll
  for (int i = 0; i < D_ / 32; ++i) {
    float v = p[lane + 32 * i];
    ss += v * v;
  }
  #pragma unroll
  for (int off = 16; off >= 1; off >>= 1)
    ss += __shfl_xor(ss, off, 32);
  if (lane == 0)
    g_inv[row] = 1.0f / fmaxf(sqrtf(ss), 1e-12f);
}

// ---------------------------------------------------------------------------
// 2) mask init
// ---------------------------------------------------------------------------
__global__ void k_init(int* __restrict__ maskInt) {
  const int c = blockIdx.x * blockDim.x + threadIdx.x;
  if (c < D_) maskInt[c] = 1;
}

// ---------------------------------------------------------------------------
// 3) WMMA GEMM: sims[b, n] ~ (t_f[b,:] . gallery[n,:]) * g_inv[n]
//    Each block: 64 gallery columns x all 128 batch rows, K=512.
//    8 waves; wave w -> M rows [16w, 16w+16), 4 N-subtiles of 16.
//    Inner loop software-pipelined: 16-K chunks double-buffered in registers
//    so 20 loads stay in flight under each group of 16 WMMAs.
// ---------------------------------------------------------------------------
__global__ void __launch_bounds__(256)
k_gemm(const float* __restrict__ t_f, const float* __restrict__ gal,
       const float* __restrict__ g_inv,
       float* __restrict__ cand_val, int* __restrict__ cand_idx) {
  __shared__ float sims[B_][TILE_N + 1];

  const int tid  = threadIdx.x;
  const int wave = tid >> 5;
  const int lane = tid & 31;
  const int half = lane >> 4;   // selects K pair per ISA A/B layout
  const int r    = lane & 15;
  const int m0   = wave * 16;
  const int nb   = blockIdx.x * TILE_N;

  v8f acc0 = {}, acc1 = {}, acc2 = {}, acc3 = {};

  // A: 16x4 f32 tile of t_f rows [m0..m0+16); lane holds (K=2h, 2h+1) of row r
  const float* arow = t_f + (size_t)(m0 + r) * D_ + 2 * half;
  // B: 4x16 f32 tile = gallery rows (columns of sims); same per-lane K pairing
  const float* b0 = gal + (size_t)(nb + 0 * 16 + r) * D_ + 2 * half;
  const float* b1 = gal + (size_t)(nb + 1 * 16 + r) * D_ + 2 * half;
  const float* b2 = gal + (size_t)(nb + 2 * 16 + r) * D_ + 2 * half;
  const float* b3 = gal + (size_t)(nb + 3 * 16 + r) * D_ + 2 * half;

  // One chunk = 16 K-values = 4 WMMA steps; operands double-buffered.
  auto loadChunk = [&](v2f (&a4)[4], v2f (&b44)[4][4], int kb) {
    #pragma unroll
    for (int s = 0; s < 4; ++s) {
      const int kk = kb + 4 * s;
      a4[s].x = arow[kk];       a4[s].y = arow[kk + 1];
      b44[0][s].x = b0[kk];     b44[0][s].y = b0[kk + 1];
      b44[1][s].x = b1[kk];     b44[1][s].y = b1[kk + 1];
      b44[2][s].x = b2[kk];     b44[2][s].y = b2[kk + 1];
      b44[3][s].x = b3[kk];     b44[3][s].y = b3[kk + 1];
    }
  };
  auto mmChunk = [&](v2f (&a4)[4], v2f (&b44)[4][4]) {
    #pragma unroll
    for (int s = 0; s < 4; ++s) {
      acc0 = __builtin_amdgcn_wmma_f32_16x16x4_f32(false, a4[s], false, b44[0][s], (short)0, acc0, false, false);
      acc1 = __builtin_amdgcn_wmma_f32_16x16x4_f32(false, a4[s], false, b44[1][s], (short)0, acc1, false, false);
      acc2 = __builtin_amdgcn_wmma_f32_16x16x4_f32(false, a4[s], false, b44[2][s], (short)0, acc2, false, false);
      acc3 = __builtin_amdgcn_wmma_f32_16x16x4_f32(false, a4[s], false, b44[3][s], (short)0, acc3, false, false);
    }
  };

  v2f aA[4], bA[4][4];
  v2f aB[4], bB[4][4];
  loadChunk(aA, bA, 0);                       // prologue
  for (int kc = 0; kc < D_; kc += 32) {
    loadChunk(aB, bB, kc + 16);               // next chunk in flight...
    __builtin_prefetch(b0 + kc + 256, 0, 1);  // ...and L2 prefetch one tile ahead
    mmChunk(aA, bA);                          // ...while consuming current
    const int kn = (kc + 32 < D_) ? (kc + 32) : 0;   // tail: harmless reload
    loadChunk(aA, bA, kn);
    mmChunk(aB, bB);
  }

  // Epilogue: scale by per-column g_inv, stage tile in LDS.
  // C/D layout: VGPR j, lanes<16 -> M=j, lanes>=16 -> M=j+8; N = lane&15.
  const float gi0 = g_inv[nb + 0 * 16 + r];
  const float gi1 = g_inv[nb + 1 * 16 + r];
  const float gi2 = g_inv[nb + 2 * 16 + r];
  const float gi3 = g_inv[nb + 3 * 16 + r];
  #pragma unroll
  for (int j = 0; j < 8; ++j) {
    const int mloc = m0 + j + 8 * half;
    sims[mloc][0 * 16 + r] = acc0[j] * gi0;
    sims[mloc][1 * 16 + r] = acc1[j] * gi1;
    sims[mloc][2 * 16 + r] = acc2[j] * gi2;
    sims[mloc][3 * 16 + r] = acc3[j] * gi3;
  }
  __syncthreads();

  // Per-row top-5 within this 64-column tile (threads 0..127, one row each).
  if (tid < B_) {
    float bvv[KTOP]; int bii[KTOP];
    #pragma unroll
    for (int i = 0; i < KTOP; ++i) { bvv[i] = -3.4e38f; bii[i] = 0x7fffffff; }
    for (int c = 0; c < TILE_N; ++c) {
      const float v = sims[tid][c];
      const int  gidx = nb + c;
      if (v > bvv[KTOP - 1] || (v == bvv[KTOP - 1] && gidx < bii[KTOP - 1])) {
        int p = KTOP - 1;
        while (p > 0 && (v > bvv[p - 1] || (v == bvv[p - 1] && gidx < bii[p - 1]))) {
          bvv[p] = bvv[p - 1]; bii[p] = bii[p - 1]; --p;
        }
        bvv[p] = v; bii[p] = gidx;
      }
    }
    const size_t base = ((size_t)tid * NUM_TILES + blockIdx.x) * KTOP;
    #pragma unroll
    for (int i = 0; i < KTOP; ++i) { cand_val[base + i] = bvv[i]; cand_idx[base + i] = bii[i]; }
  }
}

// ---------------------------------------------------------------------------
// 4) Merge per-tile candidates -> global top-5 indices per row
// ---------------------------------------------------------------------------
__global__ void k_topk_merge(const float* __restrict__ cand_val,
                             const int* __restrict__ cand_idx,
                             int* __restrict__ top_idx) {
  __shared__ float sv[256 * KTOP];
  __shared__ int   si[256 * KTOP];
  const int b = blockIdx.x, tid = threadIdx.x;
  const float* cv = cand_val + (size_t)b * NUM_TILES * KTOP;
  const int*   ci = cand_idx + (size_t)b * NUM_TILES * KTOP;

  float bv[KTOP]; int bi[KTOP];
  #pragma unroll
  for (int i = 0; i < KTOP; ++i) { bv[i] = -3.4e38f; bi[i] = 0x7fffffff; }
  for (int e = tid; e < NUM_TILES * KTOP; e += 256) {
    const float v = cv[e]; const int gidx = ci[e];
    if (v > bv[KTOP - 1] || (v == bv[KTOP - 1] && gidx < bi[KTOP - 1])) {
      int p = KTOP - 1;
      while (p > 0 && (v > bv[p - 1] || (v == bv[p - 1] && gidx < bi[p - 1]))) {
        bv[p] = bv[p - 1]; bi[p] = bi[p - 1]; --p;
      }
      bv[p] = v; bi[p] = gidx;
    }
  }
  #pragma unroll
  for (int i = 0; i < KTOP; ++i) { sv[tid * KTOP + i] = bv[i]; si[tid * KTOP + i] = bi[i]; }
  __syncthreads();

  if (tid == 0) {
    float tv[KTOP]; int ti[KTOP];
    #pragma unroll
    for (int i = 0; i < KTOP; ++i) { tv[i] = -3.4e38f; ti[i] = 0x7fffffff; }
    for (int e = 0; e < 256 * KTOP; ++e) {
      const float v = sv[e]; const int gidx = si[e];
      if (v > tv[KTOP - 1] || (v == tv[KTOP - 1] && gidx < ti[KTOP - 1])) {
        int p = KTOP - 1;
        while (p > 0 && (v > tv[p - 1] || (v == tv[p - 1] && gidx < ti[p - 1]))) {
          tv[p] = tv[p - 1]; ti[p] = ti[p - 1]; --p;
        }
        tv[p] = v; ti[p] = gidx;
      }
    }
    #pragma unroll
    for (int i = 0; i < KTOP; ++i) top_idx[b * KTOP + i] = ti[i];
  }
}

// ---------------------------------------------------------------------------
// 5) Membership: bottom-256 of |gallery[n,c] * s_f[b,c]| per (b,k) row,
//    AND-reduced into maskInt via atomics. (Row norms are positive scalars ->
//    ranking identical to the normalized reference.)
// ---------------------------------------------------------------------------
__global__ void k_member(const float* __restrict__ s_f, const float* __restrict__ gal,
                         const int* __restrict__ top_idx, int* __restrict__ maskInt) {
  __shared__ float v[D_];
  const int b = blockIdx.x / KTOP;
  const int k = blockIdx.x % KTOP;
  const int n = top_idx[b * KTOP + k];
  const int tid = threadIdx.x;

  for (int c = tid; c < D_; c += 256)
    v[c] = fabsf(gal[(size_t)n * D_ + c] * s_f[(size_t)b * D_ + c]);
  __syncthreads();

  for (int c = tid; c < D_; c += 256) {
    const float vc = v[c];
    int cnt = 0;
    for (int d = 0; d < D_; ++d)
      cnt += (v[d] < vc) || (v[d] == vc && d < c);   // stable-argsort rank
    atomicAnd(&maskInt[c], (cnt < M_KEEP) ? 1 : 0);
  }
}

// ---------------------------------------------------------------------------
// 6) mask -> float output (zero_out channels -> 0.0, else 1.0)
// ---------------------------------------------------------------------------
__global__ void k_out(const int* __restrict__ maskInt, float* __restrict__ out) {
  const int c = blockIdx.x * blockDim.x + threadIdx.x;
  if (c < D_) out[c] = maskInt[c] ? 0.0f : 1.0f;
}

// ---------------------------------------------------------------------------
extern "C" void kernel_launch(void* const* d_in, const int* in_sizes, int n_in,
                              void* d_out, int out_size, void* d_ws, size_t ws_size,
                              hipStream_t stream) {
  const float* s_f = (const float*)d_in[0];
  const float* t_f = (const float*)d_in[1];
  const float* gal = (const float*)d_in[2];
  float* out = (float*)d_out;

  // Workspace carve-up (~5.5 MB total)
  float* g_inv    = (float*)d_ws;                                   // N_
  float* cand_val = g_inv + N_;                                     // B_*NUM_TILES*KTOP
  int*   cand_idx = (int*)(cand_val + (size_t)B_ * NUM_TILES * KTOP);
  int*   top_idx  = cand_idx + (size_t)B_ * NUM_TILES * KTOP;       // B_*KTOP
  int*   maskInt  = top_idx + B_ * KTOP;                            // D_

  k_ginv<<<N_ / 8, 256, 0, stream>>>(gal, g_inv);
  k_init<<<(D_ + 255) / 256, 256, 0, stream>>>(maskInt);
  k_gemm<<<NUM_TILES, 256, 0, stream>>>(t_f, gal, g_inv, cand_val, cand_idx);
  k_topk_merge<<<B_, 256, 0, stream>>>(cand_val, cand_idx, top_idx);
  k_member<<<B_ * KTOP, 256, 0, stream>>>(s_f, gal, top_idx, maskInt);
  k_out<<<(D_ + 255) / 256, 256, 0, stream>>>(maskInt, out);
}